// TaskGating_8873402433775
// MI455X (gfx1250) — compile-verified
//
#include <hip/hip_runtime.h>
#include <hip/hip_bf16.h>
#include <math.h>

typedef __attribute__((ext_vector_type(2))) float v2f;
typedef __attribute__((ext_vector_type(8))) float v8f;

// Problem constants (from reference): B=16 S=512 E=T=768 H=1024, rows=8192
#define ROWS    8192
#define KDIM_T  768     // E == T
#define HDIM    1024
#define TWOH    2048
#define LN_EPS  1e-5f
#define MTILE   32      // rows per workgroup (2 WMMA row-fragments)

// ---------------------------------------------------------------------------
// Kernel 1: ner/topic transform.  grid = (ROWS/32, 2), block = 256 (8 waves).
// Workgroup: C[32 x 1024] = X[32 x 768] @ W[768 x 1024] (+bias, LN, ReLU).
// Wave w owns columns [w*128, w*128+128); two 16-row fragments share every
// B fragment (halves L2 traffic and vmem issue per WMMA vs 16-row tiles).
// ---------------------------------------------------------------------------
__global__ __launch_bounds__(256) void transform_kernel(
    const float* __restrict__ Xn, const float* __restrict__ Xt,
    const float* __restrict__ Wn, const float* __restrict__ Wt,
    const float* __restrict__ bn, const float* __restrict__ bt,
    const float* __restrict__ gn, const float* __restrict__ gt,
    const float* __restrict__ ben, const float* __restrict__ bet,
    float* __restrict__ comb)
{
    const int task = blockIdx.y;                 // 0 = ner, 1 = topic
    const float* __restrict__ X     = task ? Xt  : Xn;   // [8192, 768]
    const float* __restrict__ W     = task ? Wt  : Wn;   // [768, 1024]
    const float* __restrict__ bias  = task ? bt  : bn;
    const float* __restrict__ gamma = task ? gt  : gn;
    const float* __restrict__ beta  = task ? bet : ben;

    const int m0   = blockIdx.x << 5;            // 32-row tile
    const int tid  = threadIdx.x;
    const int lane = tid & 31;
    const int wave = tid >> 5;
    const int hi   = (lane >> 4) & 1;            // lanes 16..31 hold K+2 / M+8
    const int l16  = lane & 15;
    const int n0   = wave << 7;                  // 128 cols per wave

    v8f acc[2][8] = {};

    const float* __restrict__ Xr0 = X + (size_t)(m0 + l16) * KDIM_T + 2 * hi;
    const float* __restrict__ Xr1 = Xr0 + (size_t)16 * KDIM_T;

    for (int k = 0; k < KDIM_T; k += 4) {
        const v2f a0 = *(const v2f*)(Xr0 + k);   // VGPR0 = K even, VGPR1 = K odd
        const v2f a1 = *(const v2f*)(Xr1 + k);
        const float* __restrict__ Wk = W + (size_t)(k + 2 * hi) * HDIM;
#pragma unroll
        for (int t = 0; t < 8; ++t) {
            const int n = n0 + (t << 4) + l16;
            v2f b;
            b.x = Wk[n];                         // row K (+2 for hi half)
            b.y = Wk[HDIM + n];                  // row K+1
            acc[0][t] = __builtin_amdgcn_wmma_f32_16x16x4_f32(
                false, a0, false, b, (short)0, acc[0][t], false, false);
            acc[1][t] = __builtin_amdgcn_wmma_f32_16x16x4_f32(
                false, a1, false, b, (short)0, acc[1][t], false, false);
        }
    }

    // bias add + per-column gamma/beta fetch
    float gcol[8], bcol[8];
#pragma unroll
    for (int t = 0; t < 8; ++t) {
        const int n = n0 + (t << 4) + l16;
        const float bb = bias[n];
        gcol[t] = gamma[n];
        bcol[t] = beta[n];
#pragma unroll
        for (int f = 0; f < 2; ++f)
#pragma unroll
            for (int i = 0; i < 8; ++i) acc[f][t][i] += bb;
    }

    // Deterministic LayerNorm reduction: per-lane partials -> LDS -> row owners
    __shared__ float psum[256][16];
    __shared__ float psqs[256][16];
    __shared__ float mu[32], inv[32];
#pragma unroll
    for (int f = 0; f < 2; ++f)
#pragma unroll
        for (int i = 0; i < 8; ++i) {
            float s = 0.f, q = 0.f;
#pragma unroll
            for (int t = 0; t < 8; ++t) { const float v = acc[f][t][i]; s += v; q += v * v; }
            psum[tid][f * 8 + i] = s; psqs[tid][f * 8 + i] = q;
        }
    __syncthreads();
    if (tid < 32) {
        const int r  = tid;                      // row within 32-row tile
        const int rr = r & 15;
        const int off = (rr >= 8) ? 16 : 0;      // which half-wave holds it
        const int j  = ((r >> 4) << 3) + (rr & 7);
        float s = 0.f, q = 0.f;
        for (int w = 0; w < 8; ++w)
            for (int l = 0; l < 16; ++l) {
                const int idx = (w << 5) + off + l;
                s += psum[idx][j]; q += psqs[idx][j];
            }
        const float m = s * (1.0f / 1024.0f);
        const float var = q * (1.0f / 1024.0f) - m * m;
        mu[r] = m; inv[r] = rsqrtf(var + LN_EPS);
    }
    __syncthreads();

    // normalize + ReLU + store into combined
    float* __restrict__ dst = comb + (size_t)m0 * TWOH + (size_t)task * HDIM;
#pragma unroll
    for (int f = 0; f < 2; ++f)
#pragma unroll
        for (int t = 0; t < 8; ++t) {
            const int n = n0 + (t << 4) + l16;
#pragma unroll
            for (int i = 0; i < 8; ++i) {
                const int r = (f << 4) + i + (hi << 3);
                float v = (acc[f][t][i] - mu[r]) * inv[r] * gcol[t] + bcol[t];
                v = fmaxf(v, 0.0f);
                dst[(size_t)r * TWOH + n] = v;
            }
        }
}

// ---------------------------------------------------------------------------
// Kernel 2: gate MLP, only for the selected language (mask picks exactly one
// language per batch, so this is exact, 5x less work).  grid = ROWS/32.
// All 32 rows of a tile share one batch (512 rows/batch) hence one language.
// h = relu(LN(comb[32x2048] @ Wg1[lang] + bg1));  g = sigmoid(h @ Wg2 + bg2)
// out_ner = ner_t * g0, out_top = top_t * g1
// ---------------------------------------------------------------------------
__global__ __launch_bounds__(256) void gate_kernel(
    const float* __restrict__ comb,      // [8192, 2048]
    const int*   __restrict__ lang_ids,  // [16]
    const float* __restrict__ Wg1,       // [5, 2048, 1024]
    const float* __restrict__ bg1,       // [5, 1024]
    const float* __restrict__ gg,        // [5, 1024]
    const float* __restrict__ bgv,       // [5, 1024]
    const float* __restrict__ Wg2,       // [5, 1024, 2]
    const float* __restrict__ bg2,       // [5, 2]
    float* __restrict__ out_ner,         // [8192, 1024]
    float* __restrict__ out_top)         // [8192, 1024]
{
    const int m0   = blockIdx.x << 5;
    const int bidx = m0 >> 9;                    // 512 rows per batch
    const int lang = lang_ids[bidx];

    const float* __restrict__ W     = Wg1 + (size_t)lang * TWOH * HDIM;
    const float* __restrict__ bias  = bg1 + (size_t)lang * HDIM;
    const float* __restrict__ gamma = gg  + (size_t)lang * HDIM;
    const float* __restrict__ beta  = bgv + (size_t)lang * HDIM;
    const float* __restrict__ W2    = Wg2 + (size_t)lang * HDIM * 2;

    const int tid  = threadIdx.x;
    const int lane = tid & 31;
    const int wave = tid >> 5;
    const int hi   = (lane >> 4) & 1;
    const int l16  = lane & 15;
    const int n0   = wave << 7;

    v8f acc[2][8] = {};
    const float* __restrict__ Xr0 = comb + (size_t)(m0 + l16) * TWOH + 2 * hi;
    const float* __restrict__ Xr1 = Xr0 + (size_t)16 * TWOH;

    for (int k = 0; k < TWOH; k += 4) {
        const v2f a0 = *(const v2f*)(Xr0 + k);
        const v2f a1 = *(const v2f*)(Xr1 + k);
        const float* __restrict__ Wk = W + (size_t)(k + 2 * hi) * HDIM;
#pragma unroll
        for (int t = 0; t < 8; ++t) {
            const int n = n0 + (t << 4) + l16;
            v2f b;
            b.x = Wk[n];
            b.y = Wk[HDIM + n];
            acc[0][t] = __builtin_amdgcn_wmma_f32_16x16x4_f32(
                false, a0, false, b, (short)0, acc[0][t], false, false);
            acc[1][t] = __builtin_amdgcn_wmma_f32_16x16x4_f32(
                false, a1, false, b, (short)0, acc[1][t], false, false);
        }
    }

    float gcol[8], bcol[8];
#pragma unroll
    for (int t = 0; t < 8; ++t) {
        const int n = n0 + (t << 4) + l16;
        const float bb = bias[n];
        gcol[t] = gamma[n];
        bcol[t] = beta[n];
#pragma unroll
        for (int f = 0; f < 2; ++f)
#pragma unroll
            for (int i = 0; i < 8; ++i) acc[f][t][i] += bb;
    }

    __shared__ float psum[256][16];
    __shared__ float psqs[256][16];
    __shared__ float mu[32], inv[32];
#pragma unroll
    for (int f = 0; f < 2; ++f)
#pragma unroll
        for (int i = 0; i < 8; ++i) {
            float s = 0.f, q = 0.f;
#pragma unroll
            for (int t = 0; t < 8; ++t) { const float v = acc[f][t][i]; s += v; q += v * v; }
            psum[tid][f * 8 + i] = s; psqs[tid][f * 8 + i] = q;
        }
    __syncthreads();
    if (tid < 32) {
        const int r  = tid;
        const int rr = r & 15;
        const int off = (rr >= 8) ? 16 : 0;
        const int j  = ((r >> 4) << 3) + (rr & 7);
        float s = 0.f, q = 0.f;
        for (int w = 0; w < 8; ++w)
            for (int l = 0; l < 16; ++l) {
                const int idx = (w << 5) + off + l;
                s += psum[idx][j]; q += psqs[idx][j];
            }
        const float m = s * (1.0f / 1024.0f);
        const float var = q * (1.0f / 1024.0f) - m * m;
        mu[r] = m; inv[r] = rsqrtf(var + LN_EPS);
    }
    __syncthreads();

    // h = relu(LN(acc)) in place, and per-lane partial h @ Wg2
    float g0p[2][8] = {}, g1p[2][8] = {};
#pragma unroll
    for (int t = 0; t < 8; ++t) {
        const int n = n0 + (t << 4) + l16;
        const v2f w2 = *(const v2f*)(W2 + (size_t)n * 2);
#pragma unroll
        for (int f = 0; f < 2; ++f)
#pragma unroll
            for (int i = 0; i < 8; ++i) {
                const int r = (f << 4) + i + (hi << 3);
                float v = (acc[f][t][i] - mu[r]) * inv[r] * gcol[t] + bcol[t];
                v = fmaxf(v, 0.0f);
                g0p[f][i] += v * w2.x;
                g1p[f][i] += v * w2.y;
            }
    }

    // deterministic gate reduction (reuse LDS arrays)
    __syncthreads();
#pragma unroll
    for (int f = 0; f < 2; ++f)
#pragma unroll
        for (int i = 0; i < 8; ++i) {
            psum[tid][f * 8 + i] = g0p[f][i];
            psqs[tid][f * 8 + i] = g1p[f][i];
        }
    __shared__ float gate0[32], gate1[32];
    __syncthreads();
    if (tid < 32) {
        const int r  = tid;
        const int rr = r & 15;
        const int off = (rr >= 8) ? 16 : 0;
        const int j  = ((r >> 4) << 3) + (rr & 7);
        float s0 = 0.f, s1 = 0.f;
        for (int w = 0; w < 8; ++w)
            for (int l = 0; l < 16; ++l) {
                const int idx = (w << 5) + off + l;
                s0 += psum[idx][j]; s1 += psqs[idx][j];
            }
        s0 += bg2[lang * 2 + 0];
        s1 += bg2[lang * 2 + 1];
        gate0[r] = 1.0f / (1.0f + __expf(-s0));
        gate1[r] = 1.0f / (1.0f + __expf(-s1));
    }
    __syncthreads();

    // fused gating: out = combined * gate, coalesced over columns
    const float* __restrict__ src = comb + (size_t)m0 * TWOH;
    for (int idx = tid; idx < MTILE * TWOH; idx += 256) {
        const int r = idx >> 11;           // row within tile (0..31)
        const int c = idx & (TWOH - 1);    // 0..2047
        const float v = src[(size_t)r * TWOH + c];
        if (c < HDIM)
            out_ner[(size_t)(m0 + r) * HDIM + c] = v * gate0[r];
        else
            out_top[(size_t)(m0 + r) * HDIM + (c - HDIM)] = v * gate1[r];
    }
}

// ---------------------------------------------------------------------------
extern "C" void kernel_launch(void* const* d_in, const int* in_sizes, int n_in,
                              void* d_out, int out_size, void* d_ws, size_t ws_size,
                              hipStream_t stream) {
    const float* ner    = (const float*)d_in[0];
    const float* top    = (const float*)d_in[1];
    const int*   lid    = (const int*)  d_in[2];
    const float* W_ner  = (const float*)d_in[3];
    const float* b_ner  = (const float*)d_in[4];
    const float* g_ner  = (const float*)d_in[5];
    const float* be_ner = (const float*)d_in[6];
    const float* W_top  = (const float*)d_in[7];
    const float* b_top  = (const float*)d_in[8];
    const float* g_top  = (const float*)d_in[9];
    const float* be_top = (const float*)d_in[10];
    const float* Wg1    = (const float*)d_in[11];
    const float* bg1    = (const float*)d_in[12];
    const float* gg     = (const float*)d_in[13];
    const float* bg     = (const float*)d_in[14];
    const float* Wg2    = (const float*)d_in[15];
    const float* bg2    = (const float*)d_in[16];

    float* comb    = (float*)d_ws;                         // [8192, 2048]
    float* out_ner = (float*)d_out;                        // [8192, 1024]
    float* out_top = out_ner + (size_t)ROWS * HDIM;        // [8192, 1024]

    dim3 grid1(ROWS / MTILE, 2);
    transform_kernel<<<grid1, 256, 0, stream>>>(
        ner, top, W_ner, W_top, b_ner, b_top, g_ner, g_top, be_ner, be_top, comb);

    gate_kernel<<<ROWS / MTILE, 256, 0, stream>>>(
        comb, lid, Wg1, bg1, gg, bg, Wg2, bg2, out_ner, out_top);
}